// DisentangledAttention_78082505441291
// MI455X (gfx1250) — compile-verified
//
#include <hip/hip_runtime.h>
#include <math.h>

typedef __attribute__((ext_vector_type(2))) float v2f;
typedef __attribute__((ext_vector_type(8))) float v8f;

#define B_  4
#define L_  1024
#define D_  512
#define H_  8
#define DK_ 64
#define C_  8

__device__ __forceinline__ v8f wmma4(v2f a, v2f b, v8f c) {
  // D = A(16x4, f32) * B(4x16, f32) + C(16x16, f32)
  return __builtin_amdgcn_wmma_f32_16x16x4_f32(false, a, false, b, (short)0, c,
                                               false, false);
}

// CDNA5 async global->LDS copies (ASYNCcnt-tracked, no VGPR round-trip).
// %0 = LDS byte address (low 32 bits of shared-aperture flat address),
// %1 = 64-bit global address.
__device__ __forceinline__ void async_ld_b128(unsigned lds, const float* g) {
  asm volatile("global_load_async_to_lds_b128 %0, %1, off" ::"v"(lds), "v"(g)
               : "memory");
}
__device__ __forceinline__ void async_ld_b64(unsigned lds, const float* g) {
  asm volatile("global_load_async_to_lds_b64 %0, %1, off" ::"v"(lds), "v"(g)
               : "memory");
}
template <int N>
__device__ __forceinline__ void wait_async_le() {
#if __has_builtin(__builtin_amdgcn_s_wait_asynccnt)
  __builtin_amdgcn_s_wait_asynccnt(N);
#else
  asm volatile("s_wait_asynccnt %0" ::"i"(N) : "memory");
#endif
}
__device__ __forceinline__ unsigned lds_addr(const void* p) {
  return (unsigned)(size_t)p;  // low 32 bits of LDS-aperture flat addr = LDS offset
}

// ---------------------------------------------------------------------------
// Generic C = (A * W^T + bias) * scale  ; A:[M,K] row-major, W:[N,K] row-major
// Block: 256 threads (8 waves), tile 32(M) x 64(N), K step 16.
// Double-buffered async tile pipeline: 2 async instrs per wave per tile.
// ---------------------------------------------------------------------------
__global__ __launch_bounds__(256) void gemm_nt_kernel(
    const float* __restrict__ A, const float* __restrict__ W,
    const float* __restrict__ bias, float* __restrict__ out,
    int M, int N, int K, float scale) {
  __shared__ __align__(16) float As[2][32][20];
  __shared__ __align__(16) float Ws[2][64][20];
  const int tid = threadIdx.x;
  const int lane = tid & 31, wid = tid >> 5;
  const int g = lane >> 4, ml = lane & 15;
  const int mt = wid >> 2, nt = wid & 3;
  const int mbase = blockIdx.y * 32;
  const int nbase = blockIdx.x * 64;
  // per-thread async staging slots
  const int ar = tid >> 3, ac = (tid & 7) * 2;   // A: 32x16, b64 each
  const int wr = tid >> 2, wc = (tid & 3) * 4;   // W: 64x16, b128 each
  const unsigned ldsA[2] = {lds_addr(&As[0][ar][ac]), lds_addr(&As[1][ar][ac])};
  const unsigned ldsW[2] = {lds_addr(&Ws[0][wr][wc]), lds_addr(&Ws[1][wr][wc])};
  const float* gA = A + (unsigned)((mbase + ar) * K + ac);
  const float* gW = W + (unsigned)((nbase + wr) * K + wc);

  auto issue = [&](int t) {
    async_ld_b64(ldsA[t & 1], gA + t * 16);
    async_ld_b128(ldsW[t & 1], gW + t * 16);
  };

  const int NT = K / 16;
  issue(0);
  issue(1);

  v8f acc = {};
  for (int t = 0; t < NT; ++t) {
    if (t + 1 < NT)
      wait_async_le<2>();  // oldest tile (t) complete, tile t+1 in flight
    else
      wait_async_le<0>();
    __syncthreads();
    const int bf = t & 1;
#pragma unroll
    for (int ks = 0; ks < 4; ++ks) {
      int k0 = 4 * ks + 2 * g;
      v2f a, b;
      a.x = As[bf][16 * mt + ml][k0];
      a.y = As[bf][16 * mt + ml][k0 + 1];
      b.x = Ws[bf][16 * nt + ml][k0];
      b.y = Ws[bf][16 * nt + ml][k0 + 1];
      acc = wmma4(a, b, acc);
    }
    __syncthreads();  // all waves done reading buffer bf
    if (t + 2 < NT) issue(t + 2);
  }
#pragma unroll
  for (int v = 0; v < 8; ++v) {
    int m = mbase + 16 * mt + 8 * g + v;
    int n = nbase + 16 * nt + ml;
    out[(unsigned)(m * N + n)] = (acc[v] + bias[n]) * scale;
  }
}

// ---------------------------------------------------------------------------
// Clustering: probs [B,H,L,C] for q (src=0) and k (src=1), + KL partials.
// Grid: 256 blocks (src*B*H*chunks), 256 threads (one token each).
// ---------------------------------------------------------------------------
__global__ __launch_bounds__(256) void cluster_kernel(
    const float* __restrict__ query, const float* __restrict__ key,
    const float* __restrict__ mu, const float* __restrict__ log_var,
    const float* __restrict__ log_prior,
    float* __restrict__ probs_q, float* __restrict__ probs_k,
    float* __restrict__ kl_part) {
  const int bid = blockIdx.x;
  const int chunk = bid & 3;
  const int h = (bid >> 2) & 7;
  const int b = (bid >> 5) & 3;
  const int src = bid >> 7;
  const int tid = threadIdx.x;
  const int l = chunk * 256 + tid;

  __shared__ float mu_s[C_ * DK_];
  __shared__ float iv_s[C_ * DK_];
  __shared__ float lv_s[C_ * DK_];
  __shared__ float lvsum_s[C_], s2_s[C_], logp_s[C_], prior_s[C_];
  __shared__ float red[256];

  const float ALV = -4.605170185988091f;  // log(0.01)
  const float AVAR = expf(ALV);

  for (int i = tid; i < C_ * DK_; i += 256) {
    float lv = log_var[h * C_ * DK_ + i];
    mu_s[i] = mu[h * C_ * DK_ + i];
    lv_s[i] = lv;
    iv_s[i] = expf(-lv);
  }
  __syncthreads();
  if (tid < C_) {
    float ls = 0.f, s2 = 0.f;
    for (int d = 0; d < DK_; ++d) {
      ls += lv_s[tid * DK_ + d];
      s2 += AVAR * iv_s[tid * DK_ + d] + lv_s[tid * DK_ + d];
    }
    lvsum_s[tid] = ls;
    s2_s[tid] = s2;
    float mx = -3.4e38f;
    for (int c = 0; c < C_; ++c) mx = fmaxf(mx, log_prior[h * C_ + c]);
    float se = 0.f;
    for (int c = 0; c < C_; ++c) se += expf(log_prior[h * C_ + c] - mx);
    float lse = mx + logf(se);
    logp_s[tid] = log_prior[h * C_ + tid] - lse;
    prior_s[tid] = expf(logp_s[tid]);
  }
  __syncthreads();

  const float* x =
      (src ? key : query) + (unsigned)((b * L_ + l) * D_ + h * DK_);
  float mse[C_];
#pragma unroll
  for (int c = 0; c < C_; ++c) mse[c] = 0.f;
  for (int d4 = 0; d4 < DK_ / 4; ++d4) {
    float4 xq = *(const float4*)(x + 4 * d4);
    float xs[4] = {xq.x, xq.y, xq.z, xq.w};
#pragma unroll
    for (int j = 0; j < 4; ++j) {
      float xv = xs[j];
      int d = 4 * d4 + j;
#pragma unroll
      for (int c = 0; c < C_; ++c) {
        float diff = xv - mu_s[c * DK_ + d];
        mse[c] += diff * diff * iv_s[c * DK_ + d];
      }
    }
  }
  float lp[C_];
  float mx = -3.4e38f;
#pragma unroll
  for (int c = 0; c < C_; ++c) {
    lp[c] = -0.5f * mse[c] - 64.0f * 3.14159265358979323846f -
            0.5f * lvsum_s[c] + logp_s[c];
    mx = fmaxf(mx, lp[c]);
  }
  float se = 0.f;
#pragma unroll
  for (int c = 0; c < C_; ++c) se += expf(lp[c] - mx);
  float lz = mx + logf(se);
  float* pout =
      (src ? probs_k : probs_q) + (unsigned)(((b * H_ + h) * L_ + l) * C_);
  float t1 = 0.f, t2 = 0.f;
#pragma unroll
  for (int c = 0; c < C_; ++c) {
    float lpr = lp[c] - lz;
    float pr = expf(lpr);
    pout[c] = pr;
    t1 += prior_s[c] * (logp_s[c] - lpr);
    t2 += pr * (mse[c] + s2_s[c]);
  }
  red[tid] = t1 + 0.5f * t2;
  __syncthreads();
  for (int s = 128; s > 0; s >>= 1) {
    if (tid < s) red[tid] += red[tid + s];
    __syncthreads();
  }
  if (tid == 0) kl_part[bid] = red[0];
}

// ---------------------------------------------------------------------------
// div_loss partials: sum over tile of ((P P^T) - I)^2 * wmask.
// Grid: 512 blocks (b,h,qchunk of 64 rows), 128 threads (4 waves x 16 rows).
// ---------------------------------------------------------------------------
__global__ __launch_bounds__(128) void div_kernel(
    const float* __restrict__ probs, float* __restrict__ part) {
  const int bid = blockIdx.x;
  const int qc = bid & 15;
  const int h = (bid >> 4) & 7;
  const int b = bid >> 7;
  const int tid = threadIdx.x;
  const int lane = tid & 31, wid = tid >> 5;
  const int g = lane >> 4, ml = lane & 15;
  const int qbase = qc * 64 + wid * 16;
  const float* P = probs + (unsigned)((b * H_ + h) * L_ * C_);
  v2f a0, a1;
  {
    const float* row = P + (unsigned)((qbase + ml) * C_);
    a0.x = row[2 * g];
    a0.y = row[2 * g + 1];
    a1.x = row[4 + 2 * g];
    a1.y = row[4 + 2 * g + 1];
  }
  float wsum = 0.f;
  for (int ct = 0; ct < 64; ++ct) {
    int cb = ct * 16;
    const float* col = P + (unsigned)((cb + ml) * C_);
    v2f b0, b1;
    b0.x = col[2 * g];
    b0.y = col[2 * g + 1];
    b1.x = col[4 + 2 * g];
    b1.y = col[4 + 2 * g + 1];
    v8f acc = {};
    acc = wmma4(a0, b0, acc);
    acc = wmma4(a1, b1, acc);
#pragma unroll
    for (int v = 0; v < 8; ++v) {
      int m = qbase + 8 * g + v;
      int n = cb + ml;
      float tgt = (m == n) ? 1.f : 0.f;
      float w = (m == n) ? 1.25f : 0.75f;
      float d = acc[v] - tgt;
      wsum += d * d * w;
    }
  }
  for (int msk = 16; msk > 0; msk >>= 1) wsum += __shfl_xor(wsum, msk, 32);
  __shared__ float sred[4];
  if (lane == 0) sred[wid] = wsum;
  __syncthreads();
  if (tid == 0) part[bid] = sred[0] + sred[1] + sred[2] + sred[3];
}

// ---------------------------------------------------------------------------
// Fused masked attention (flash-style online softmax).
// Grid: 512 blocks (b,h,64-row q chunk), 128 threads (4 waves x 16 q rows).
// K/V tiles double-buffered through the CDNA5 async global->LDS pipeline:
// 4 async instrs per wave per tile; s_wait_asynccnt 4 retires the oldest tile.
// ---------------------------------------------------------------------------
__global__ __launch_bounds__(128) void attn_kernel(
    const float* __restrict__ qp, const float* __restrict__ kp,
    const float* __restrict__ vp, const float* __restrict__ probs_q,
    const float* __restrict__ probs_k, const float* __restrict__ pad,
    float* __restrict__ outp) {
  const int bid = blockIdx.x;
  const int qc = bid & 15;
  const int h = (bid >> 4) & 7;
  const int b = bid >> 7;
  const int tid = threadIdx.x;
  const int lane = tid & 31, wid = tid >> 5;
  const int g = lane >> 4, ml = lane & 15;
  const int qbase = qc * 64 + wid * 16;

  __shared__ __align__(16) float Kt[2][16][68];  // 68-float stride: 16B rows,
  __shared__ __align__(16) float Vt[2][16][68];  // conflict-free 16-lane gathers
  __shared__ float Pks[2][16][9];
  __shared__ float pads[2][16];
  __shared__ float pws[4][16][17];

  // per-thread async staging slot: row r8, 8 columns starting at cc
  const int r8 = tid >> 3;
  const int cc = (tid & 7) * 8;
  const unsigned ldsK[2] = {lds_addr(&Kt[0][r8][cc]), lds_addr(&Kt[1][r8][cc])};
  const unsigned ldsV[2] = {lds_addr(&Vt[0][r8][cc]), lds_addr(&Vt[1][r8][cc])};
  const float* gkRow = kp + (unsigned)((b * L_ + r8) * D_ + h * DK_ + cc);
  const float* gvRow = vp + (unsigned)((b * L_ + r8) * D_ + h * DK_ + cc);
  const int pr = tid >> 3, pc = tid & 7;
  const float* gpRow = probs_k + (unsigned)(((b * H_ + h) * L_ + pr) * C_ + pc);

  auto issue = [&](int t) {
    const int bf = t & 1;
    const float* gk = gkRow + t * 16 * D_;
    const float* gv = gvRow + t * 16 * D_;
    async_ld_b128(ldsK[bf], gk);
    async_ld_b128(ldsK[bf] + 16, gk + 4);
    async_ld_b128(ldsV[bf], gv);
    async_ld_b128(ldsV[bf] + 16, gv + 4);
    Pks[bf][pr][pc] = gpRow[t * 16 * C_];
    if (tid < 16) pads[bf][tid] = pad[(unsigned)(b * L_ + t * 16 + tid)];
  };

  v2f qa[16];
  {
    const float* qrow = qp + (unsigned)((b * L_ + qbase + ml) * D_ + h * DK_);
#pragma unroll
    for (int s = 0; s < 16; ++s) {
      qa[s].x = qrow[4 * s + 2 * g];
      qa[s].y = qrow[4 * s + 2 * g + 1];
    }
  }
  v2f pqa[2];
  {
    const float* prow =
        probs_q + (unsigned)(((b * H_ + h) * L_ + qbase + ml) * C_);
#pragma unroll
    for (int s = 0; s < 2; ++s) {
      pqa[s].x = prow[4 * s + 2 * g];
      pqa[s].y = prow[4 * s + 2 * g + 1];
    }
  }

  v8f zero = {};
  v8f acc[4];
#pragma unroll
  for (int t4 = 0; t4 < 4; ++t4) acc[t4] = zero;
  float mrow[8], zrow[8];
#pragma unroll
  for (int v = 0; v < 8; ++v) {
    mrow[v] = -3.0e38f;
    zrow[v] = 0.f;
  }

  issue(0);
  issue(1);

  for (int kt = 0; kt < 64; ++kt) {
    if (kt + 1 < 64)
      wait_async_le<4>();  // tile kt landed; tile kt+1 still in flight
    else
      wait_async_le<0>();
    __syncthreads();  // tile kt visible to all waves
    const int bf = kt & 1;

    // similarity tile (K = 8) -> cluster mask
    v8f sim = zero;
#pragma unroll
    for (int s = 0; s < 2; ++s) {
      v2f bb;
      bb.x = Pks[bf][ml][4 * s + 2 * g];
      bb.y = Pks[bf][ml][4 * s + 2 * g + 1];
      sim = wmma4(pqa[s], bb, sim);
    }
    float cm[8];
    v8f sc;
    float padv = pads[bf][ml];
#pragma unroll
    for (int v = 0; v < 8; ++v) {
      float c = 1.f - sim[v] + padv;
      c = fminf(fmaxf(c, 0.f), 1.f);
      cm[v] = c;
      sc[v] = -10000.f * c;
    }
    // scores += Q K^T (K = 64)
#pragma unroll
    for (int s = 0; s < 16; ++s) {
      v2f bb;
      bb.x = Kt[bf][ml][4 * s + 2 * g];
      bb.y = Kt[bf][ml][4 * s + 2 * g + 1];
      sc = wmma4(qa[s], bb, sc);
    }
    // online softmax row update (rows live in 16-lane groups)
    float pw[8];
#pragma unroll
    for (int v = 0; v < 8; ++v) {
      float tmax = sc[v];
      tmax = fmaxf(tmax, __shfl_xor(tmax, 1, 32));
      tmax = fmaxf(tmax, __shfl_xor(tmax, 2, 32));
      tmax = fmaxf(tmax, __shfl_xor(tmax, 4, 32));
      tmax = fmaxf(tmax, __shfl_xor(tmax, 8, 32));
      float mn = fmaxf(mrow[v], tmax);
      float scale = expf(mrow[v] - mn);
      float pv = expf(sc[v] - mn);
      float ts = pv;
      ts += __shfl_xor(ts, 1, 32);
      ts += __shfl_xor(ts, 2, 32);
      ts += __shfl_xor(ts, 4, 32);
      ts += __shfl_xor(ts, 8, 32);
      zrow[v] = zrow[v] * scale + ts;
      mrow[v] = mn;
      pw[v] = pv * (1.f - cm[v]);
#pragma unroll
      for (int t4 = 0; t4 < 4; ++t4) acc[t4][v] *= scale;
    }
    // transpose weighted probs (C/D layout -> A layout) via LDS
    __syncthreads();
#pragma unroll
    for (int v = 0; v < 8; ++v) pws[wid][8 * g + v][ml] = pw[v];
    __syncthreads();
    // acc += P V  (K = 16 keys, N = 64 dims)
#pragma unroll
    for (int s = 0; s < 4; ++s) {
      v2f pa;
      pa.x = pws[wid][ml][4 * s + 2 * g];
      pa.y = pws[wid][ml][4 * s + 2 * g + 1];
#pragma unroll
      for (int t4 = 0; t4 < 4; ++t4) {
        v2f vb;
        vb.x = Vt[bf][4 * s + 2 * g][16 * t4 + ml];
        vb.y = Vt[bf][4 * s + 2 * g + 1][16 * t4 + ml];
        acc[t4] = wmma4(pa, vb, acc[t4]);
      }
    }
    __syncthreads();  // all waves done reading buffer bf
    if (kt + 2 < 64) issue(kt + 2);
  }
#pragma unroll
  for (int t4 = 0; t4 < 4; ++t4) {
#pragma unroll
    for (int v = 0; v < 8; ++v) {
      int m = qbase + 8 * g + v;
      int n = 16 * t4 + ml;
      outp[(unsigned)((b * L_ + m) * D_ + h * DK_ + n)] = acc[t4][v] / zrow[v];
    }
  }
}

// ---------------------------------------------------------------------------
// Deterministic finalize of kl_loss / div_loss into d_out tail.
// ---------------------------------------------------------------------------
__global__ void finalize_kernel(const float* __restrict__ kl_part,
                                const float* __restrict__ div_q,
                                const float* __restrict__ div_k,
                                float* __restrict__ tail) {
  int b = threadIdx.x;
  if (b >= 4) return;
  const float ALV = -4.605170185988091f;
  const float KCONST = -0.5f * (1.f + ALV) * 64.f;
  float s = 0.f;
  for (int src = 0; src < 2; ++src)
    for (int h = 0; h < 8; ++h)
      for (int ch = 0; ch < 4; ++ch)
        s += kl_part[src * 128 + b * 32 + h * 4 + ch];
  float kl = s / (8.f * 1024.f) + 2.f * KCONST;
  float dv = 0.f;
  for (int h = 0; h < 8; ++h)
    for (int qc = 0; qc < 16; ++qc)
      dv += div_q[b * 128 + h * 16 + qc] + div_k[b * 128 + h * 16 + qc];
  dv /= 8.f * 1024.f * 1024.f;
  tail[b] = kl;
  tail[4 + b] = dv;
}

extern "C" void kernel_launch(void* const* d_in, const int* in_sizes, int n_in,
                              void* d_out, int out_size, void* d_ws,
                              size_t ws_size, hipStream_t stream) {
  const float* query = (const float*)d_in[0];
  const float* key = (const float*)d_in[1];
  const float* value = (const float*)d_in[2];
  const float* pad = (const float*)d_in[3];
  const float* Wq = (const float*)d_in[4];
  const float* bq = (const float*)d_in[5];
  const float* Wk = (const float*)d_in[6];
  const float* bk = (const float*)d_in[7];
  const float* Wv = (const float*)d_in[8];
  const float* bv = (const float*)d_in[9];
  const float* Wo = (const float*)d_in[10];
  const float* bo = (const float*)d_in[11];
  const float* mu = (const float*)d_in[12];
  const float* lv = (const float*)d_in[13];
  const float* lprior = (const float*)d_in[14];
  float* out = (float*)d_out;

  float* ws = (float*)d_ws;
  const size_t MD = (size_t)B_ * L_ * D_;  // 2,097,152
  float* qProj = ws;
  float* kProj = qProj + MD;
  float* vProj = kProj + MD;
  float* aOut = vProj + MD;
  float* probsQ = aOut + MD;
  float* probsK = probsQ + (size_t)B_ * H_ * L_ * C_;
  float* klPart = probsK + (size_t)B_ * H_ * L_ * C_;
  float* divQ = klPart + 256;
  float* divK = divQ + 512;

  cluster_kernel<<<256, 256, 0, stream>>>(query, key, mu, lv, lprior, probsQ,
                                          probsK, klPart);
  div_kernel<<<512, 128, 0, stream>>>(probsQ, divQ);
  div_kernel<<<512, 128, 0, stream>>>(probsK, divK);

  dim3 gg(512 / 64, 4096 / 32);
  gemm_nt_kernel<<<gg, 256, 0, stream>>>(query, Wq, bq, qProj, 4096, 512, 512,
                                         0.125f);
  gemm_nt_kernel<<<gg, 256, 0, stream>>>(key, Wk, bk, kProj, 4096, 512, 512,
                                         1.0f);
  gemm_nt_kernel<<<gg, 256, 0, stream>>>(value, Wv, bv, vProj, 4096, 512, 512,
                                         1.0f);

  attn_kernel<<<512, 128, 0, stream>>>(qProj, kProj, vProj, probsQ, probsK,
                                       pad, aOut);

  gemm_nt_kernel<<<gg, 256, 0, stream>>>(aOut, Wo, bo, out, 4096, 512, 512,
                                         1.0f);

  finalize_kernel<<<1, 32, 0, stream>>>(klPart, divQ, divK, out + MD);
}